// OCHits2ShowersLayer_26173530702550
// MI455X (gfx1250) — compile-verified
//
#include <hip/hip_runtime.h>

#define NHITS 200000
#define NSEG  6250          // NHITS / 32, also the assigned-bitmask word count
#define MAX_SHOWERS 2048
#define BATCH 16
#define BETA_TH 0.3f
#define BLOCK_HITS 800      // 25 segments * 32 hits, 50 tiles of 16
#define NBLOCKS_B 250       // 250 * 800 = 200000
#define SEG_PER_BLOCK 25
#define TILES_PER_BLOCK 50
#define ROUNDS 132

typedef __attribute__((ext_vector_type(2))) float v2f;
typedef __attribute__((ext_vector_type(8))) float v8f;

struct WS {
  unsigned segmax[NSEG];   // max beta-key of unassigned eligible hits per 32-hit segment
  unsigned bits[NSEG];     // assigned bitmask, 1 bit per hit
  unsigned keys[NHITS];    // eligible ? float_bits(beta) : 0
  int done;
  int k;                   // showers created so far
  int nseed;               // seeds in current batch
  int pad;
  float sx[BATCH], sy[BATCH], sz[BATCH], snorm[BATCH], sr2[BATCH];
  int   sidx[BATCH], ssid[BATCH];
};

__device__ __forceinline__ unsigned ballot32(bool p) {
#if __has_builtin(__builtin_amdgcn_ballot_w32)
  return __builtin_amdgcn_ballot_w32(p);
#else
  return (unsigned)__ballot((int)p);
#endif
}

__device__ __forceinline__ void wait_async0() {
#if __has_builtin(__builtin_amdgcn_s_wait_asynccnt)
  __builtin_amdgcn_s_wait_asynccnt(0);
#else
  asm volatile("s_wait_asynccnt 0x0" ::: "memory");
#endif
}

// ---------------- init: outputs, keys, bitmask ----------------
__global__ void k_init_a(const float* __restrict__ beta, int* __restrict__ out,
                         WS* __restrict__ ws) {
  int i = blockIdx.x * blockDim.x + threadIdx.x;
  if (i < NHITS) {
    out[i] = -1;            // assoc_idx
    out[NHITS + i] = -1;    // shower_id
    float b = beta[i];
    ws->keys[i] = (b > BETA_TH) ? __float_as_uint(b) : 0u;
  }
  if (i < NSEG) ws->bits[i] = 0u;
}

// ---------------- init: segment maxima + batch state ----------------
__global__ void k_init_b(WS* __restrict__ ws, int* __restrict__ out) {
  int s = blockIdx.x * blockDim.x + threadIdx.x;
  if (s < NSEG) {
    unsigned m = 0;
    int base = s * 32;
    for (int j = 0; j < 32; j++) {
      unsigned kk = ws->keys[base + j];
      if (kk > m) m = kk;
    }
    ws->segmax[s] = m;
  }
  if (s == 0) {
    ws->done = 0; ws->k = 0; ws->nseed = 0;
    out[2 * NHITS] = 0;
  }
}

// ---------------- per-round seed selection (1 block, 1024 threads) ----------------
__global__ __launch_bounds__(1024) void k_select(const float* __restrict__ cc,
                                                 const float* __restrict__ dist,
                                                 int* __restrict__ out,
                                                 WS* __restrict__ ws) {
  __shared__ unsigned long long red[32];
  __shared__ int s_exit;
  const int t = threadIdx.x;
  const int lane = t & 31;
  const int wv = t >> 5;

  if (t == 0) {
    int d = ws->done;
    s_exit = d;
    ws->nseed = 0;
    if (d) out[2 * NHITS] = ws->k;
  }
  __syncthreads();
  if (s_exit) return;

  for (;;) {
    // ---- parallel argmax over segment maxima (tie -> lowest segment) ----
    unsigned long long best = 0;
    for (int seg = t; seg < NSEG; seg += 1024) {
      unsigned key = ws->segmax[seg];
      if (key) {
        unsigned long long p =
            ((unsigned long long)key << 32) | (unsigned)(NSEG - 1 - seg);
        if (p > best) best = p;
      }
    }
    // wave32 max-reduce
    for (int off = 16; off > 0; off >>= 1) {
      unsigned long long o = __shfl_xor(best, off, 32);
      if (o > best) best = o;
    }
    if (lane == 0) red[wv] = best;
    __syncthreads();
    if (wv == 0) {
      best = red[lane];                 // 32 waves -> 32 partials
      for (int off = 16; off > 0; off >>= 1) {
        unsigned long long o = __shfl_xor(best, off, 32);
        if (o > best) best = o;
      }
    }

    if (t == 0) {
      unsigned long long b = best;
      int exitloop = 0;
      if (b == 0ull) {
        ws->done = 1;      // no eligible unassigned hits remain
        exitloop = 1;
      } else {
        unsigned key = (unsigned)(b >> 32);
        int seg = NSEG - 1 - (int)(unsigned)(b & 0xffffffffu);
        int base = seg * 32;
        unsigned wbits = ws->bits[seg];
        int s = base;
        for (int j = 0; j < 32; j++) {
          int idx = base + j;
          if (ws->keys[idx] == key && !((wbits >> j) & 1u)) { s = idx; break; }
        }
        float x = cc[3 * s + 0], y = cc[3 * s + 1], z = cc[3 * s + 2];
        int ns = ws->nseed;
        int cov = -1;
        for (int j = 0; j < ns; j++) {
          float dx = x - ws->sx[j], dy = y - ws->sy[j], dz = z - ws->sz[j];
          float d2 = dx * dx + dy * dy + dz * dz;
          if (d2 < ws->sr2[j]) { cov = j; break; }
        }
        ws->bits[seg] = wbits | (1u << (s - base));   // mark assigned
        if (cov >= 0) {
          out[s] = ws->sidx[cov];
          out[NHITS + s] = ws->ssid[cov];
        } else {
          int kk = ws->k;
          ws->sx[ns] = x; ws->sy[ns] = y; ws->sz[ns] = z;
          ws->snorm[ns] = x * x + y * y + z * z;
          float r = dist[s];
          ws->sr2[ns] = r * r;
          ws->sidx[ns] = s; ws->ssid[ns] = kk;
          out[s] = s; out[NHITS + s] = kk;
          ws->k = kk + 1;
          ws->nseed = ns + 1;
          if (kk + 1 >= MAX_SHOWERS) { ws->done = 1; exitloop = 1; }
          else if (ns + 1 >= BATCH) exitloop = 1;
        }
        // refresh this segment's max
        unsigned m = 0;
        unsigned nb = ws->bits[seg];
        for (int j = 0; j < 32; j++)
          if (!((nb >> j) & 1u)) {
            unsigned kk2 = ws->keys[base + j];
            if (kk2 > m) m = kk2;
          }
        ws->segmax[seg] = m;
      }
      s_exit = exitloop;
    }
    __syncthreads();
    if (s_exit) break;
  }

  if (t == 0) {
    for (int j = ws->nseed; j < BATCH; j++) {  // pad unused columns: never match
      ws->sx[j] = 0.f; ws->sy[j] = 0.f; ws->sz[j] = 0.f;
      ws->snorm[j] = 0.f; ws->sr2[j] = -1.0f;
      ws->sidx[j] = -1; ws->ssid[j] = -1;
    }
    out[2 * NHITS] = ws->k;
  }
}

// ---------------- per-round assignment: WMMA 16 hits x 16 seeds ----------------
__global__ __launch_bounds__(256) void k_assign(const float* __restrict__ cc,
                                                int* __restrict__ out,
                                                WS* __restrict__ ws) {
  const int ns = ws->nseed;
  if (ns == 0) return;

  __shared__ __align__(16) float lcoord[BLOCK_HITS * 3];  // 9.6 KB staged coords
  __shared__ int s_any;

  const int tid = threadIdx.x;
  const int lane = tid & 31;
  const int wv = tid >> 5;
  const int col = lane & 15;
  const bool lo = lane < 16;
  const int blockBase = blockIdx.x * BLOCK_HITS;

  // ---- early-out if every hit in this block is already assigned ----
  if (tid == 0) s_any = 0;
  __syncthreads();
  if (tid < SEG_PER_BLOCK) {
    if (ws->bits[blockIdx.x * SEG_PER_BLOCK + tid] != 0xffffffffu)
      atomicOr(&s_any, 1);
  }
  __syncthreads();
  if (!s_any) return;

  // ---- async-stage this block's coords into LDS (ASYNCcnt path) ----
  {
    const char* gbase = (const char*)(cc + (size_t)blockBase * 3);
    const unsigned lbase = (unsigned)(size_t)&lcoord[0];
    for (int c = tid; c < (BLOCK_HITS * 3 * 4) / 16; c += 256) {  // 600 x 16B
      const char* gp = gbase + (size_t)c * 16;
      unsigned lp = lbase + (unsigned)c * 16;
      asm volatile("global_load_async_to_lds_b128 %0, %1, off"
                   :
                   : "v"(lp), "v"(gp)
                   : "memory");
    }
    wait_async0();
  }
  __syncthreads();

  // B operand (4x16): rows are (-2sx, -2sy, -2sz, 1); lane%16 = column N
  v2f Bm;
  Bm.x = lo ? (-2.0f * ws->sx[col]) : (-2.0f * ws->sz[col]);  // K0 | K2
  Bm.y = lo ? (-2.0f * ws->sy[col]) : 1.0f;                   // K1 | K3
  // per-lane threshold: D = |h|^2 - 2 h.s ;  D < r2 - |s|^2  <=>  d2 < r2
  const float thr = ws->sr2[col] - ws->snorm[col];

  for (int tile = wv; tile < TILES_PER_BLOCK; tile += 8) {
    const int hbase = blockBase + tile * 16;
    const unsigned word = ws->bits[hbase >> 5];
    const unsigned mask16 = (word >> (((hbase >> 4) & 1) ? 16 : 0)) & 0xffffu;
    if (mask16 == 0xffffu) continue;   // whole tile already assigned

    const int lh = tile * 16 + col;    // local hit index
    const float x = lcoord[3 * lh + 0];
    const float y = lcoord[3 * lh + 1];
    const float z = lcoord[3 * lh + 2];
    const float nrm = x * x + y * y + z * z;

    // A operand (16x4): row M = lane%16, K = (x, y, z, |h|^2)
    v2f Am;
    Am.x = lo ? x : z;     // K0 | K2
    Am.y = lo ? y : nrm;   // K1 | K3

    v8f Cz = {};           // zero accumulator (inline SRC2 = 0)
    v8f Dm = __builtin_amdgcn_wmma_f32_16x16x4_f32(
        false, Am, false, Bm, (short)0, Cz, false, false);

    unsigned m[8];
#pragma unroll
    for (int v = 0; v < 8; v++)
      m[v] = ballot32(Dm[v] < thr);   // wave-uniform per-row coverage masks

    // wave-uniform scalar resolution: first covering seed (lowest col) per row
    const unsigned avail = ~mask16 & 0xffffu;
#pragma unroll
    for (int v = 0; v < 8; v++) {
      const unsigned lom = m[v] & 0xffffu;
      const unsigned him = m[v] >> 16;
      if (((avail >> v) & 1u) && lom) {
        const int j = __builtin_ctz(lom);
        if (lane == 0) {
          const int hr = hbase + v;
          out[hr] = ws->sidx[j];
          out[NHITS + hr] = ws->ssid[j];
          atomicOr(&ws->bits[hr >> 5], 1u << (hr & 31));
        }
      }
      if (((avail >> (v + 8)) & 1u) && him) {
        const int j = __builtin_ctz(him);
        if (lane == 0) {
          const int hr = hbase + v + 8;
          out[hr] = ws->sidx[j];
          out[NHITS + hr] = ws->ssid[j];
          atomicOr(&ws->bits[hr >> 5], 1u << (hr & 31));
        }
      }
    }
  }
  __syncthreads();

  // refresh this block's 25 segment maxima
  if (tid < SEG_PER_BLOCK) {
    const int seg = blockIdx.x * SEG_PER_BLOCK + tid;
    const unsigned w = atomicOr(&ws->bits[seg], 0u);  // coherent read
    const int base = seg * 32;
    unsigned mx = 0;
    for (int j = 0; j < 32; j++)
      if (!((w >> j) & 1u)) {
        unsigned kk = ws->keys[base + j];
        if (kk > mx) mx = kk;
      }
    ws->segmax[seg] = mx;
  }
}

extern "C" void kernel_launch(void* const* d_in, const int* in_sizes, int n_in,
                              void* d_out, int out_size, void* d_ws, size_t ws_size,
                              hipStream_t stream) {
  (void)in_sizes; (void)n_in; (void)out_size;
  if (ws_size < sizeof(WS)) return;

  const float* cc   = (const float*)d_in[0];  // (N,3) f32
  const float* beta = (const float*)d_in[1];  // (N,1) f32
  const float* dist = (const float*)d_in[2];  // (N,1) f32
  int* out = (int*)d_out;                     // assoc[N] | sid[N] | k
  WS* ws = (WS*)d_ws;

  hipLaunchKernelGGL(k_init_a, dim3((NHITS + 255) / 256), dim3(256), 0, stream,
                     beta, out, ws);
  hipLaunchKernelGGL(k_init_b, dim3((NSEG + 255) / 256), dim3(256), 0, stream,
                     ws, out);
  for (int r = 0; r < ROUNDS; r++) {
    hipLaunchKernelGGL(k_select, dim3(1), dim3(1024), 0, stream, cc, dist, out, ws);
    hipLaunchKernelGGL(k_assign, dim3(NBLOCKS_B), dim3(256), 0, stream, cc, out, ws);
  }
}